// _ECELoss_55662776156556
// MI455X (gfx1250) — compile-verified
//
#include <hip/hip_runtime.h>
#include <hip/hip_bf16.h>
#include <cstdint>
#include <math.h>

#define NCLS 100
#define NBINS 15
#define ROWS_PER_TILE 32
#define WAVES_PER_BLOCK 2
#define THREADS (WAVES_PER_BLOCK * 32)
#define TILE_FLOATS (ROWS_PER_TILE * NCLS)      // 3200 floats = 12800 B
#define TILE_BYTES (TILE_FLOATS * 4)
#define MAX_BLOCKS 2048

// ---- CDNA5 async global->LDS helpers (gfx1250) -----------------------------

// Stage one 12800-byte tile (32 contiguous rows of 100 floats) into LDS.
// Each lane copies chunks (lane + 32*i)*16B, i = 0..24  -> 25 async B128 ops.
// INST_OFFSET is added to BOTH the global and the LDS address, so a single
// base pair per lane + immediate offsets covers the whole tile.
__device__ __forceinline__ void issue_tile_load(const float* gbase,
                                                uint32_t lds_base,
                                                int lane) {
    uint64_t g = (uint64_t)(uintptr_t)gbase + (uint64_t)lane * 16u;
    uint32_t l = lds_base + (uint32_t)lane * 16u;
    asm volatile(
        "global_load_async_to_lds_b128 %0, %1, off\n\t"
        "global_load_async_to_lds_b128 %0, %1, off offset:512\n\t"
        "global_load_async_to_lds_b128 %0, %1, off offset:1024\n\t"
        "global_load_async_to_lds_b128 %0, %1, off offset:1536\n\t"
        "global_load_async_to_lds_b128 %0, %1, off offset:2048\n\t"
        "global_load_async_to_lds_b128 %0, %1, off offset:2560\n\t"
        "global_load_async_to_lds_b128 %0, %1, off offset:3072\n\t"
        "global_load_async_to_lds_b128 %0, %1, off offset:3584\n\t"
        "global_load_async_to_lds_b128 %0, %1, off offset:4096\n\t"
        "global_load_async_to_lds_b128 %0, %1, off offset:4608\n\t"
        "global_load_async_to_lds_b128 %0, %1, off offset:5120\n\t"
        "global_load_async_to_lds_b128 %0, %1, off offset:5632\n\t"
        "global_load_async_to_lds_b128 %0, %1, off offset:6144\n\t"
        "global_load_async_to_lds_b128 %0, %1, off offset:6656\n\t"
        "global_load_async_to_lds_b128 %0, %1, off offset:7168\n\t"
        "global_load_async_to_lds_b128 %0, %1, off offset:7680\n\t"
        "global_load_async_to_lds_b128 %0, %1, off offset:8192\n\t"
        "global_load_async_to_lds_b128 %0, %1, off offset:8704\n\t"
        "global_load_async_to_lds_b128 %0, %1, off offset:9216\n\t"
        "global_load_async_to_lds_b128 %0, %1, off offset:9728\n\t"
        "global_load_async_to_lds_b128 %0, %1, off offset:10240\n\t"
        "global_load_async_to_lds_b128 %0, %1, off offset:10752\n\t"
        "global_load_async_to_lds_b128 %0, %1, off offset:11264\n\t"
        "global_load_async_to_lds_b128 %0, %1, off offset:11776\n\t"
        "global_load_async_to_lds_b128 %0, %1, off offset:12288\n\t"
        :: "v"(l), "v"(g) : "memory");
}

__device__ __forceinline__ void wait_async_le(void) {   // first batch (25) done
    asm volatile("s_wait_asynccnt 0x19" ::: "memory");
}
__device__ __forceinline__ void wait_async_0(void) {
    asm volatile("s_wait_asynccnt 0x0" ::: "memory");
}
__device__ __forceinline__ void wait_ds_0(void) {       // LDS reads/atomics done
    asm volatile("s_wait_dscnt 0x0" ::: "memory");
}

// ---- kernels ---------------------------------------------------------------

__global__ void ece_init_ws(float* ws) {
    if (threadIdx.x < 3 * NBINS) ws[threadIdx.x] = 0.0f;
}

__global__ void __launch_bounds__(THREADS)
ece_main(const float* __restrict__ logits, const int* __restrict__ labels,
         float* __restrict__ ws, int ntiles) {
    __shared__ __align__(16) float tiles[WAVES_PER_BLOCK * 2 * TILE_FLOATS];
    __shared__ float s_bins[3 * NBINS];

    const int lane = threadIdx.x & 31;
    const int wave = threadIdx.x >> 5;

    if (threadIdx.x < 3 * NBINS) s_bins[threadIdx.x] = 0.0f;
    __syncthreads();

    float* buf0p = &tiles[wave * 2 * TILE_FLOATS];
    uint32_t ldsbuf0 = (uint32_t)(uintptr_t)(void*)buf0p;
    uint32_t ldsbuf1 = ldsbuf0 + TILE_BYTES;

    const int totalWaves = gridDim.x * WAVES_PER_BLOCK;
    int t = blockIdx.x * WAVES_PER_BLOCK + wave;
    int cur = 0;

    if (t < ntiles)
        issue_tile_load(logits + (size_t)t * TILE_FLOATS, ldsbuf0, lane);

    for (; t < ntiles; t += totalWaves) {
        const int nxt = t + totalWaves;
        if (nxt < ntiles) {
            issue_tile_load(logits + (size_t)nxt * TILE_FLOATS,
                            cur ? ldsbuf0 : ldsbuf1, lane);
            wait_async_le();          // batch for tile t complete (in-order)
        } else {
            wait_async_0();
        }

        // lane processes one full row -> exact first-match argmax semantics
        const float4* row = (const float4*)(&tiles[wave * 2 * TILE_FLOATS +
                                                   cur * TILE_FLOATS +
                                                   lane * NCLS]);
        float mx = -INFINITY;
        int am = 0;
#pragma unroll
        for (int j = 0; j < NCLS / 4; ++j) {
            float4 v = row[j];
            if (v.x > mx) { mx = v.x; am = 4 * j + 0; }
            if (v.y > mx) { mx = v.y; am = 4 * j + 1; }
            if (v.z > mx) { mx = v.z; am = 4 * j + 2; }
            if (v.w > mx) { mx = v.w; am = 4 * j + 3; }
        }
        float s = 0.0f;
#pragma unroll
        for (int j = 0; j < NCLS / 4; ++j) {
            float4 v = row[j];
            s += __expf(v.x - mx) + __expf(v.y - mx) +
                 __expf(v.z - mx) + __expf(v.w - mx);
        }
        const float conf = 1.0f / s;           // max(softmax) = 1/sum(exp(x-max))
        const int rowidx = t * ROWS_PER_TILE + lane;
        const float acc = (am == labels[rowidx]) ? 1.0f : 0.0f;

        int bin = (int)ceilf(conf * (float)NBINS) - 1;
        bin = bin < 0 ? 0 : (bin > NBINS - 1 ? NBINS - 1 : bin);

        atomicAdd(&s_bins[bin], 1.0f);
        atomicAdd(&s_bins[NBINS + bin], conf);
        atomicAdd(&s_bins[2 * NBINS + bin], acc);

        wait_ds_0();   // all LDS reads/atomics retired before buffer reuse
        cur ^= 1;
    }

    __syncthreads();
    if (threadIdx.x < 3 * NBINS)
        atomicAdd(&ws[threadIdx.x], s_bins[threadIdx.x]);
}

// remainder rows (N % 32) — at most 31 rows, plain global reads
__global__ void ece_tail(const float* __restrict__ logits,
                         const int* __restrict__ labels,
                         float* __restrict__ ws, int start, int nrem) {
    int i = (int)threadIdx.x;
    if (i >= nrem) return;
    const int r = start + i;
    const float* row = logits + (size_t)r * NCLS;
    float mx = -INFINITY;
    int am = 0;
    for (int j = 0; j < NCLS; ++j) {
        float v = row[j];
        if (v > mx) { mx = v; am = j; }
    }
    float s = 0.0f;
    for (int j = 0; j < NCLS; ++j) s += __expf(row[j] - mx);
    const float conf = 1.0f / s;
    const float acc = (am == labels[r]) ? 1.0f : 0.0f;
    int bin = (int)ceilf(conf * (float)NBINS) - 1;
    bin = bin < 0 ? 0 : (bin > NBINS - 1 ? NBINS - 1 : bin);
    atomicAdd(&ws[bin], 1.0f);
    atomicAdd(&ws[NBINS + bin], conf);
    atomicAdd(&ws[2 * NBINS + bin], acc);
}

__global__ void ece_final(const float* __restrict__ ws, float* __restrict__ out,
                          float invN) {
    if (threadIdx.x == 0) {
        float ece = 0.0f;
        for (int b = 0; b < NBINS; ++b) {
            const float cnt = ws[b];
            const float sc = cnt > 1.0f ? cnt : 1.0f;
            const float avg_conf = ws[NBINS + b] / sc;
            const float acc_in = ws[2 * NBINS + b] / sc;
            out[1 + b] = avg_conf;
            out[1 + NBINS + b] = acc_in;
            ece += fabsf(avg_conf - acc_in) * cnt * invN;
        }
        out[0] = ece;
    }
}

// ---- entry -----------------------------------------------------------------

extern "C" void kernel_launch(void* const* d_in, const int* in_sizes, int n_in,
                              void* d_out, int out_size, void* d_ws,
                              size_t ws_size, hipStream_t stream) {
    const float* logits = (const float*)d_in[0];
    const int* labels = (const int*)d_in[1];
    float* ws = (float*)d_ws;
    float* out = (float*)d_out;

    const int N = in_sizes[1];                 // number of samples
    const int ntiles = N / ROWS_PER_TILE;
    const int rem = N - ntiles * ROWS_PER_TILE;

    ece_init_ws<<<1, 64, 0, stream>>>(ws);

    if (ntiles > 0) {
        int blocks = (ntiles + WAVES_PER_BLOCK - 1) / WAVES_PER_BLOCK;
        if (blocks > MAX_BLOCKS) blocks = MAX_BLOCKS;
        ece_main<<<blocks, THREADS, 0, stream>>>(logits, labels, ws, ntiles);
    }
    if (rem > 0) {
        ece_tail<<<1, 32, 0, stream>>>(logits, labels, ws,
                                       ntiles * ROWS_PER_TILE, rem);
    }
    ece_final<<<1, 32, 0, stream>>>(ws, out, 1.0f / (float)N);
}